// TPNMetric_19396072309314
// MI455X (gfx1250) — compile-verified
//
#include <hip/hip_runtime.h>

typedef __attribute__((ext_vector_type(2))) float v2f;
typedef __attribute__((ext_vector_type(8))) float v8f;

#define NN    4096
#define CC    256
#define NWAY  16
#define NSUP  16
#define TOPK  20
#define ALPHA_C 0.99f
#define EPSF  2.2204460492503131e-16f
#define BNEPS 1e-5f
#define CG_ITERS 200
#define KSLICES 16
#define KSLICE_LEN (NN / KSLICES)   // 256

// ---------------- pooling: x[i][c] = mean over 7x7 ----------------
__global__ void k_pool(const float* __restrict__ emb, float* __restrict__ x) {
  int idx = blockIdx.x * 256 + threadIdx.x;   // N*C = 1048576 threads exactly
  const float* p = emb + (size_t)idx * 49;
  float s = 0.f;
#pragma unroll
  for (int t = 0; t < 49; ++t) s += p[t];
  x[idx] = s * (1.0f / 49.0f);
}

// ---------------- h = x @ w1 + b1 via WMMA f32 16x16x4 ----------------
__global__ void k_gemm1(const float* __restrict__ x, const float* __restrict__ w1,
                        const float* __restrict__ b1, float* __restrict__ h) {
  int lane = threadIdx.x & 31;
  int half = lane >> 4, lr = lane & 15;
  int mtile = blockIdx.y, ntile = blockIdx.x;
  const float* xr = x + (size_t)(mtile * 16 + lr) * CC;
  int col = ntile * 16 + lr;
  v8f acc = {0.f,0.f,0.f,0.f,0.f,0.f,0.f,0.f};
  for (int k0 = 0; k0 < CC; k0 += 4) {
    int ka = k0 + half * 2;
    v2f a = { xr[ka], xr[ka + 1] };
    v2f b = { w1[(size_t)ka * CC + col], w1[(size_t)(ka + 1) * CC + col] };
    acc = __builtin_amdgcn_wmma_f32_16x16x4_f32(false, a, false, b, (short)0, acc, false, false);
  }
  int rowb = mtile * 16 + half * 8;
  float bias = b1[col];
#pragma unroll
  for (int r = 0; r < 8; ++r) h[(size_t)(rowb + r) * CC + col] = acc[r] + bias;
}

// ---------------- BN batch stats per channel ----------------
__global__ void k_bnstats(const float* __restrict__ h, float* __restrict__ mu,
                          float* __restrict__ var) {
  __shared__ float s1[256], s2[256];
  int j = blockIdx.x;
  float a = 0.f, b = 0.f;
  for (int i = threadIdx.x; i < NN; i += 256) {
    float v = h[(size_t)i * CC + j];
    a += v; b += v * v;
  }
  s1[threadIdx.x] = a; s2[threadIdx.x] = b;
  __syncthreads();
  for (int st = 128; st > 0; st >>= 1) {
    if (threadIdx.x < st) { s1[threadIdx.x] += s1[threadIdx.x + st]; s2[threadIdx.x] += s2[threadIdx.x + st]; }
    __syncthreads();
  }
  if (threadIdx.x == 0) {
    float m = s1[0] / (float)NN;
    mu[j] = m;
    var[j] = s2[0] / (float)NN - m * m;
  }
}

// ---------------- BN normalize + ReLU (in place) ----------------
__global__ void k_bnrelu(float* __restrict__ h, const float* __restrict__ mu,
                         const float* __restrict__ var, const float* __restrict__ gamma,
                         const float* __restrict__ beta) {
  int i = blockIdx.x, j = threadIdx.x;
  size_t o = (size_t)i * CC + j;
  float v = (h[o] - mu[j]) * rsqrtf(var[j] + BNEPS) * gamma[j] + beta[j];
  h[o] = v > 0.f ? v : 0.f;
}

// ---------------- sigma = sigmoid(h @ w2 + b2) ----------------
__global__ void k_sigma(const float* __restrict__ h, const float* __restrict__ w2,
                        const float* __restrict__ b2, float* __restrict__ sigma) {
  __shared__ float red[256];
  int i = blockIdx.x;
  red[threadIdx.x] = h[(size_t)i * CC + threadIdx.x] * w2[threadIdx.x];
  __syncthreads();
  for (int st = 128; st > 0; st >>= 1) {
    if (threadIdx.x < st) red[threadIdx.x] += red[threadIdx.x + st];
    __syncthreads();
  }
  if (threadIdx.x == 0) {
    float z = red[0] + b2[0];
    sigma[i] = 1.0f / (1.0f + expf(-z));
  }
}

// ---------------- e = x/(sigma+eps); sq[i]=||e_i||^2 ----------------
__global__ void k_escale(const float* __restrict__ x, const float* __restrict__ sigma,
                         float* __restrict__ e, float* __restrict__ sq) {
  __shared__ float red[256];
  int i = blockIdx.x;
  float inv = 1.0f / (sigma[i] + EPSF);
  size_t o = (size_t)i * CC + threadIdx.x;
  float v = x[o] * inv;
  e[o] = v;
  red[threadIdx.x] = v * v;
  __syncthreads();
  for (int st = 128; st > 0; st >>= 1) {
    if (threadIdx.x < st) red[threadIdx.x] += red[threadIdx.x + st];
    __syncthreads();
  }
  if (threadIdx.x == 0) sq[i] = red[0];
}

// ---------------- W = exp(-d2/2), Gram via WMMA f32 16x16x4 ----------------
__global__ void k_gram(const float* __restrict__ e, const float* __restrict__ sq,
                       float* __restrict__ W) {
  int lane = threadIdx.x & 31;
  int half = lane >> 4, lr = lane & 15;
  int mtile = blockIdx.y, ntile = blockIdx.x;
  const float* Em = e + (size_t)(mtile * 16 + lr) * CC;
  const float* En = e + (size_t)(ntile * 16 + lr) * CC;
  v8f acc = {0.f,0.f,0.f,0.f,0.f,0.f,0.f,0.f};
  for (int k0 = 0; k0 < CC; k0 += 4) {
    int ka = k0 + half * 2;
    v2f a = { Em[ka], Em[ka + 1] };
    v2f b = { En[ka], En[ka + 1] };
    acc = __builtin_amdgcn_wmma_f32_16x16x4_f32(false, a, false, b, (short)0, acc, false, false);
  }
  int rowb = mtile * 16 + half * 8;
  int col = ntile * 16 + lr;
  float sqc = sq[col];
#pragma unroll
  for (int r = 0; r < 8; ++r) {
    int row = rowb + r;
    float d2 = (sq[row] + sqc - 2.0f * acc[r]) * (1.0f / (float)CC);
    W[(size_t)row * NN + col] = expf(-0.5f * d2);
  }
}

// ---------------- per-row top-20, scatter symmetric mask ----------------
__global__ void k_topk(const float* __restrict__ W, unsigned char* __restrict__ mask) {
  __shared__ float row[NN];
  __shared__ float rv[256];
  __shared__ int ri[256];
  int i = blockIdx.x;
  for (int j = threadIdx.x; j < NN; j += 256) row[j] = W[(size_t)i * NN + j];
  __syncthreads();
  for (int t = 0; t < TOPK; ++t) {
    float best = -__builtin_huge_valf();
    int bi = NN;
    for (int j = threadIdx.x; j < NN; j += 256) {
      float v = row[j];
      if (v > best) { best = v; bi = j; }
    }
    rv[threadIdx.x] = best; ri[threadIdx.x] = bi;
    __syncthreads();
    for (int st = 128; st > 0; st >>= 1) {
      if (threadIdx.x < st) {
        float v2 = rv[threadIdx.x + st]; int i2 = ri[threadIdx.x + st];
        if (v2 > rv[threadIdx.x] || (v2 == rv[threadIdx.x] && i2 < ri[threadIdx.x])) {
          rv[threadIdx.x] = v2; ri[threadIdx.x] = i2;
        }
      }
      __syncthreads();
    }
    if (threadIdx.x == 0) {
      int j = ri[0];
      mask[(size_t)i * NN + j] = 1;
      mask[(size_t)j * NN + i] = 1;
      row[j] = -__builtin_huge_valf();
    }
    __syncthreads();
  }
}

// ---------------- apply mask in place + row degrees ----------------
__global__ void k_maskdeg(float* __restrict__ W, const unsigned char* __restrict__ mask,
                          float* __restrict__ Dg) {
  __shared__ float red[256];
  int i = blockIdx.x;
  float s = 0.f;
  for (int j = threadIdx.x; j < NN; j += 256) {
    size_t o = (size_t)i * NN + j;
    float v = mask[o] ? W[o] : 0.0f;
    W[o] = v; s += v;
  }
  red[threadIdx.x] = s;
  __syncthreads();
  for (int st = 128; st > 0; st >>= 1) {
    if (threadIdx.x < st) red[threadIdx.x] += red[threadIdx.x + st];
    __syncthreads();
  }
  if (threadIdx.x == 0) Dg[i] = red[0];
}

__global__ void k_dvec(const float* __restrict__ Dg, float* __restrict__ dvec) {
  int i = blockIdx.x * 256 + threadIdx.x;
  dvec[i] = sqrtf(1.0f / (Dg[i] + EPSF));
}

// ---------------- A = I - alpha * d_i d_j Wm + eps (in place on W) --------
__global__ void k_abuild(float* __restrict__ W, const float* __restrict__ dvec) {
  size_t idx = (size_t)blockIdx.x * 256 + threadIdx.x;
  int i = (int)(idx >> 12);     // /4096
  int j = (int)(idx & 4095);
  float v = (i == j ? 1.0f : 0.0f) - ALPHA_C * dvec[i] * dvec[j] * W[idx] + EPSF;
  W[idx] = v;
}

// ---------------- y: support one-hot ----------------
__global__ void k_yset(const int* __restrict__ elabel, float* __restrict__ y) {
  int i = blockIdx.x * 256 + threadIdx.x;  // 4096
  int pos = i & 255;                        // i % (ns+nq)
  if (pos < NSUP) y[(size_t)i * NWAY + elabel[i]] = 1.0f;
}

// ---- AP partials: K split 16-way for memory-level parallelism (WMMA) ----
__global__ void k_matvec_part(const float* __restrict__ A, const float* __restrict__ P,
                              float* __restrict__ part) {
  int lane = threadIdx.x & 31;
  int half = lane >> 4, lr = lane & 15;
  int mtile = blockIdx.y;              // 0..255
  int ks = blockIdx.x;                 // 0..15
  const float* Ar = A + (size_t)(mtile * 16 + lr) * NN + (size_t)ks * KSLICE_LEN;
  const float* Pk = P + (size_t)ks * KSLICE_LEN * NWAY;
  v8f acc = {0.f,0.f,0.f,0.f,0.f,0.f,0.f,0.f};
  for (int k0 = 0; k0 < KSLICE_LEN; k0 += 4) {
    int ka = k0 + half * 2;
    v2f a = { Ar[ka], Ar[ka + 1] };
    v2f b = { Pk[(size_t)ka * NWAY + lr], Pk[(size_t)(ka + 1) * NWAY + lr] };
    acc = __builtin_amdgcn_wmma_f32_16x16x4_f32(false, a, false, b, (short)0, acc, false, false);
  }
  int rowb = mtile * 16 + half * 8;
  float* out = part + (size_t)ks * NN * NWAY;
#pragma unroll
  for (int r = 0; r < 8; ++r) out[(size_t)(rowb + r) * NWAY + lr] = acc[r];
}

// ---- deterministic fixed-order reduction of the 16 K-slice partials ----
__global__ void k_apreduce(const float* __restrict__ part, float* __restrict__ AP) {
  int idx = blockIdx.x * 256 + threadIdx.x;   // 65536
  float s = 0.f;
#pragma unroll
  for (int ks = 0; ks < KSLICES; ++ks) s += part[(size_t)ks * NN * NWAY + idx];
  AP[idx] = s;
}

// ---------------- per-column dot of two (N,16) matrices ----------------
__global__ void k_dotcol(const float* __restrict__ a, const float* __restrict__ b,
                         float* __restrict__ out) {
  __shared__ float red[256];
  int j = blockIdx.x;  // 0..15
  float s = 0.f;
  for (int i = threadIdx.x; i < NN; i += 256)
    s += a[(size_t)i * NWAY + j] * b[(size_t)i * NWAY + j];
  red[threadIdx.x] = s;
  __syncthreads();
  for (int st = 128; st > 0; st >>= 1) {
    if (threadIdx.x < st) red[threadIdx.x] += red[threadIdx.x + st];
    __syncthreads();
  }
  if (threadIdx.x == 0) out[j] = red[0];
}

// ---------------- CG: X += alpha P; R -= alpha AP ----------------
__global__ void k_updxr(float* __restrict__ X, float* __restrict__ R,
                        const float* __restrict__ P, const float* __restrict__ AP,
                        const float* __restrict__ rr, const float* __restrict__ pAp) {
  int idx = blockIdx.x * 256 + threadIdx.x;  // 65536
  int j = idx & 15;
  float num = rr[j], den = pAp[j];
  float alpha = (den > 0.f && num > 0.f) ? num / den : 0.0f;
  X[idx] += alpha * P[idx];
  R[idx] -= alpha * AP[idx];
}

// ---------------- CG: P = R + beta P ----------------
__global__ void k_updp(float* __restrict__ P, const float* __restrict__ R,
                       const float* __restrict__ rrnew, const float* __restrict__ rrold) {
  int idx = blockIdx.x * 256 + threadIdx.x;  // 65536
  int j = idx & 15;
  float den = rrold[j];
  float beta = (den > 0.f) ? rrnew[j] / den : 0.0f;
  P[idx] = R[idx] + beta * P[idx];
}

// ---------------- final output: F floats, then elabel as floats ----------
__global__ void k_writeout(const float* __restrict__ X, const int* __restrict__ elabel,
                           float* __restrict__ out) {
  int idx = blockIdx.x * 256 + threadIdx.x;  // 69632
  if (idx < NN * NWAY) out[idx] = X[idx];
  else out[idx] = (float)elabel[idx - NN * NWAY];
}

extern "C" void kernel_launch(void* const* d_in, const int* in_sizes, int n_in,
                              void* d_out, int out_size, void* d_ws, size_t ws_size,
                              hipStream_t stream) {
  const float* emb    = (const float*)d_in[0];
  const int*   elabel = (const int*)d_in[1];
  const float* w1     = (const float*)d_in[3];
  const float* b1     = (const float*)d_in[4];
  const float* gamma  = (const float*)d_in[5];
  const float* beta   = (const float*)d_in[6];
  const float* w2     = (const float*)d_in[7];
  const float* b2     = (const float*)d_in[8];

  char* ws = (char*)d_ws;
  float* x  = (float*)(ws);                         // 4 MB
  float* h  = (float*)(ws + (4ull  << 20));         // 4 MB
  float* e  = (float*)(ws + (8ull  << 20));         // 4 MB
  float* W  = (float*)(ws + (12ull << 20));         // 64 MB (becomes A)
  unsigned char* mask = (unsigned char*)(ws + (76ull << 20)); // 16 MB
  float* part = (float*)(ws + (92ull << 20));       // 4 MB (16 K-slice partials)
  size_t base = 96ull << 20;
  float* sq    = (float*)(ws + base); base += NN * 4;
  float* sigma = (float*)(ws + base); base += NN * 4;
  float* Dg    = (float*)(ws + base); base += NN * 4;
  float* dvec  = (float*)(ws + base); base += NN * 4;
  float* mu    = (float*)(ws + base); base += CC * 4;
  float* var   = (float*)(ws + base); base += CC * 4;
  float* y     = (float*)(ws + base); base += NN * NWAY * 4;
  float* X     = (float*)(ws + base); base += NN * NWAY * 4;
  float* R     = (float*)(ws + base); base += NN * NWAY * 4;
  float* P     = (float*)(ws + base); base += NN * NWAY * 4;
  float* AP    = (float*)(ws + base); base += NN * NWAY * 4;
  float* rrA   = (float*)(ws + base); base += 64;
  float* rrB   = (float*)(ws + base); base += 64;
  float* pAp   = (float*)(ws + base); base += 64;

  // feature pipeline
  k_pool<<<(NN * CC) / 256, 256, 0, stream>>>(emb, x);
  k_gemm1<<<dim3(CC / 16, NN / 16), 32, 0, stream>>>(x, w1, b1, h);
  k_bnstats<<<CC, 256, 0, stream>>>(h, mu, var);
  k_bnrelu<<<NN, CC, 0, stream>>>(h, mu, var, gamma, beta);
  k_sigma<<<NN, CC, 0, stream>>>(h, w2, b2, sigma);
  k_escale<<<NN, CC, 0, stream>>>(x, sigma, e, sq);

  // affinity + sparsify + normalize
  k_gram<<<dim3(NN / 16, NN / 16), 32, 0, stream>>>(e, sq, W);
  hipMemsetAsync(mask, 0, (size_t)NN * NN, stream);
  k_topk<<<NN, 256, 0, stream>>>(W, mask);
  k_maskdeg<<<NN, 256, 0, stream>>>(W, mask, Dg);
  k_dvec<<<NN / 256, 256, 0, stream>>>(Dg, dvec);
  k_abuild<<<(NN * NN) / 256, 256, 0, stream>>>(W, dvec);

  // labels + CG init
  hipMemsetAsync(y, 0, (size_t)NN * NWAY * 4, stream);
  k_yset<<<NN / 256, 256, 0, stream>>>(elabel, y);
  hipMemsetAsync(X, 0, (size_t)NN * NWAY * 4, stream);
  hipMemcpyAsync(R, y, (size_t)NN * NWAY * 4, hipMemcpyDeviceToDevice, stream);
  hipMemcpyAsync(P, y, (size_t)NN * NWAY * 4, hipMemcpyDeviceToDevice, stream);
  k_dotcol<<<NWAY, 256, 0, stream>>>(R, R, rrA);

  // conjugate gradient, 16 RHS in lockstep
  float* rr_old = rrA;
  float* rr_new = rrB;
  for (int it = 0; it < CG_ITERS; ++it) {
    k_matvec_part<<<dim3(KSLICES, NN / 16), 32, 0, stream>>>(W, P, part);
    k_apreduce<<<(NN * NWAY) / 256, 256, 0, stream>>>(part, AP);
    k_dotcol<<<NWAY, 256, 0, stream>>>(P, AP, pAp);
    k_updxr<<<(NN * NWAY) / 256, 256, 0, stream>>>(X, R, P, AP, rr_old, pAp);
    k_dotcol<<<NWAY, 256, 0, stream>>>(R, R, rr_new);
    k_updp<<<(NN * NWAY) / 256, 256, 0, stream>>>(P, R, rr_new, rr_old);
    float* t = rr_old; rr_old = rr_new; rr_new = t;
  }

  k_writeout<<<(NN * NWAY + NN) / 256, 256, 0, stream>>>(X, elabel, (float*)d_out);
}